// FrankWolfePolicyImprovement_56495999811647
// MI455X (gfx1250) — compile-verified
//
#include <hip/hip_runtime.h>
#include <math.h>

// Problem constants (match reference)
constexpr int Ncol = 32;   // n
constexpr int Mrow = 24;   // m
constexpr int NSl  = 4;    // soft-constraint slacks
constexpr int Hd   = 256;  // critic hidden dim
constexpr int ITERS = 200;
constexpr float ALPHA = 0.1f;

typedef float v2f __attribute__((ext_vector_type(2)));
typedef float v8f __attribute__((ext_vector_type(8)));

// ---------------------------------------------------------------------------
// Factorized ADMM solver: one 64-thread block per problem.
//   Precompute:  Minv = (diag(P) + G^T G)^-1,  K = Minv G^T,  L = G K,
//                u0 = Minv (-q),  v0 = G u0
//   Iterate:     w = z - y;  Gu = v0 + L w;  z = min(Gu + y, h);  y += Gu - z
//   Final:       u = u0 + K w
// ANCHOR=true : D=36,R=60, q=[-x_raw;0], P=diag(1..1,2,2,2,2) -> out = u[:32]
// ANCHOR=false: D=32,R=56, q=-g, P=0 -> out = (1-a)*x_feas + a*u
// ---------------------------------------------------------------------------
template <int D, int R, bool ANCHOR>
__global__ void __launch_bounds__(64)
admm_kernel(const float* __restrict__ vin,    // x_raw (anchor) or g (lmo): [P,32]
            const float* __restrict__ Aall,   // [P,24,32]
            const float* __restrict__ ball,   // [P,24]
            const float* __restrict__ xfeas,  // [P,32] (lmo only; null for anchor)
            float* __restrict__ out)          // [P,32]
{
  const int p   = blockIdx.x;
  const int tid = threadIdx.x;
  constexpr int T = 64;

  __shared__ float sG[R][D + 1];          // constraint matrix
  __shared__ float sM[D][2 * D + 2];      // Gauss-Jordan augmented workspace
  __shared__ float sK[D][R + 1];          // Minv G^T
  __shared__ float sL[R][R + 1];          // G Minv G^T   (R+1 coprime w/ 64 banks)
  __shared__ float sFac[D];
  __shared__ float sh[R];
  __shared__ __align__(16) float sw[2][R];  // double-buffered w = z - y
  __shared__ float su0[D];
  __shared__ float sv0[R];
  __shared__ float svin[D];               // -q

  const float* Ap = Aall + (size_t)p * Mrow * Ncol;
  const float* bp = ball + (size_t)p * Mrow;

  // ---- build G, h, -q ----------------------------------------------------
  for (int idx = tid; idx < R * D; idx += T) {
    int i = idx / D, j = idx - i * D;
    float v = 0.f;
    if (ANCHOR) {
      if (i < Mrow) {                       // [A_hard 0] / [A_soft -I]
        if (j < Ncol) v = Ap[i * Ncol + j];
        else if (i >= Mrow - NSl && (i - (Mrow - NSl)) == (j - Ncol)) v = -1.f;
      } else if (i < Mrow + NSl) {          // [0 -I]  (slack >= 0)
        if (j >= Ncol && (i - Mrow) == (j - Ncol)) v = -1.f;
      } else {                              // [-I 0]  (x >= LOWER)
        if (j < Ncol && (i - (Mrow + NSl)) == j) v = -1.f;
      }
    } else {
      if (i < Mrow) { if (j < Ncol) v = Ap[i * Ncol + j]; }
      else if ((i - Mrow) == j) v = -1.f;   // -I (c >= LOWER)
    }
    sG[i][j] = v;
  }
  for (int i = tid; i < R; i += T) sh[i] = (i < Mrow) ? bp[i] : 0.f;  // LOWER==0
  for (int i = tid; i < D; i += T)
    svin[i] = (i < Ncol) ? vin[(size_t)p * Ncol + i] : 0.f;           // -q
  __syncthreads();

  // ---- M = diag(P) + G^T G, augmented with I -----------------------------
  for (int idx = tid; idx < D * D; idx += T) {
    int i = idx / D, j = idx - i * D;
    float acc = 0.f;
    if (i == j) acc = ANCHOR ? ((i < Ncol) ? 1.f : 2.f) : 0.f;  // P_diag
    for (int k = 0; k < R; ++k) acc += sG[k][i] * sG[k][j];
    sM[i][j]     = acc;
    sM[i][D + j] = (i == j) ? 1.f : 0.f;
  }
  __syncthreads();

  // ---- Gauss-Jordan inverse (SPD: no pivoting needed) --------------------
  for (int col = 0; col < D; ++col) {
    float ip = 1.0f / sM[col][col];   // prev iter ended with a barrier
    __syncthreads();                  // all read pivot before scaling
    for (int j = tid; j < 2 * D; j += T) sM[col][j] *= ip;
    for (int i = tid; i < D; i += T) sFac[i] = sM[i][col];
    __syncthreads();
    for (int idx = tid; idx < D * 2 * D; idx += T) {
      int row = idx / (2 * D), j = idx - row * 2 * D;
      if (row != col) sM[row][j] -= sFac[row] * sM[col][j];
    }
    __syncthreads();
  }

  // ---- K = Minv G^T ------------------------------------------------------
  for (int idx = tid; idx < D * R; idx += T) {
    int i = idx / R, k = idx - i * R;
    float acc = 0.f;
    for (int j = 0; j < D; ++j) acc += sM[i][D + j] * sG[k][j];
    sK[i][k] = acc;
  }
  // ---- u0 = Minv (-q) ----------------------------------------------------
  for (int i = tid; i < D; i += T) {
    float acc = 0.f;
    for (int j = 0; j < D; ++j) acc += sM[i][D + j] * svin[j];
    su0[i] = acc;
  }
  __syncthreads();
  // ---- L = G K, v0 = G u0 ------------------------------------------------
  for (int idx = tid; idx < R * R; idx += T) {
    int t = idx / R, k = idx - t * R;
    float acc = 0.f;
    for (int j = 0; j < D; ++j) acc += sG[t][j] * sK[j][k];
    sL[t][k] = acc;
  }
  for (int t = tid; t < R; t += T) {
    float acc = 0.f;
    for (int j = 0; j < D; ++j) acc += sG[t][j] * su0[j];
    sv0[t] = acc;
  }
  __syncthreads();

  // ---- cache this thread's L row in registers (invariant over 200 iters) -
  float Lrow[R];
  float zt = 0.f, yt = 0.f, htv = 0.f, v0t = 0.f;
  if (tid < R) {
    htv = sh[tid];
    v0t = sv0[tid];
    #pragma unroll
    for (int k = 0; k < R; ++k) Lrow[k] = sL[tid][k];
  }

  // ---- ADMM iterations (rho = 1); 1 barrier/iter via double buffering ----
  for (int it = 0; it < ITERS; ++it) {
    const int buf = it & 1;
    if (tid < R) sw[buf][tid] = zt - yt;
    __syncthreads();
    if (tid < R) {
      const float4* w4 = (const float4*)sw[buf];
      float acc0 = v0t, acc1 = 0.f;
      #pragma unroll
      for (int kq = 0; kq < R / 4; ++kq) {
        float4 wv = w4[kq];
        acc0 += Lrow[4 * kq + 0] * wv.x;
        acc1 += Lrow[4 * kq + 1] * wv.y;
        acc0 += Lrow[4 * kq + 2] * wv.z;
        acc1 += Lrow[4 * kq + 3] * wv.w;
      }
      float acc = acc0 + acc1;
      float zn = acc + yt;
      zt = fminf(zn, htv);
      yt = yt + acc - zt;
    }
    // next iteration writes the other buffer: no second barrier needed
  }
  {
    const int buf = ITERS & 1;
    if (tid < R) sw[buf][tid] = zt - yt;
    __syncthreads();

    // ---- u = u0 + K w ; write result -------------------------------------
    if (tid < Ncol) {
      float acc = su0[tid];
      #pragma unroll 4
      for (int k = 0; k < R; ++k) acc += sK[tid][k] * sw[buf][k];
      if (ANCHOR) {
        out[(size_t)p * Ncol + tid] = acc;
      } else {
        out[(size_t)p * Ncol + tid] =
            (1.f - ALPHA) * xfeas[(size_t)p * Ncol + tid] + ALPHA * acc;
      }
    }
  }
}

// ---------------------------------------------------------------------------
// Critic gradient via CDNA5 WMMA (V_WMMA_F32_16X16X4_F32):
//   z = x W1 + b1 ; s = (1 - tanh(z)^2) * w2 ; g = s W1^T
// One wave per 16-row strip; 2 waves (64 threads) per block.
// A-frag layout (16x4 f32): lane<16 holds M=lane,K={0,1}; lane>=16 K={2,3}.
// C/D layout: VGPR v = row (v + 8*(lane>>4)), col = lane&15.
// Branchless sech^2 via HW reciprocal:
//   e = exp(2z) (z clamped to +/-40);  1 - tanh(z)^2 = 4e / (e+1)^2
// ---------------------------------------------------------------------------
__global__ void __launch_bounds__(64)
critic_grad_kernel(const float* __restrict__ xfeas,  // [P,32]
                   const float* __restrict__ W1,     // [32,256]
                   const float* __restrict__ b1,     // [256]
                   const float* __restrict__ w2,     // [256]
                   float* __restrict__ g)            // [P,32]
{
  __shared__ float s[2][16][Hd + 1];  // per-wave activation-grad strip
  const int lane = threadIdx.x & 31;
  const int wave = threadIdx.x >> 5;
  const int row0 = (blockIdx.x * 2 + wave) * 16;
  const int half = lane >> 4;   // 0: K={0,1}, 1: K={2,3}
  const int lid  = lane & 15;

  // ---- hoist GEMM1 A-fragments (depend on kt only, not ht) ---------------
  v2f afrag[Ncol / 4];
  #pragma unroll
  for (int kt = 0; kt < Ncol / 4; ++kt) {
    int k0 = kt * 4 + half * 2;
    afrag[kt].x = xfeas[(size_t)(row0 + lid) * Ncol + k0];
    afrag[kt].y = xfeas[(size_t)(row0 + lid) * Ncol + k0 + 1];
  }

  // ---- GEMM1: z = x @ W1, fused activation-grad -> s in LDS --------------
  for (int ht = 0; ht < Hd / 16; ++ht) {
    v8f acc = {};
    #pragma unroll
    for (int kt = 0; kt < Ncol / 4; ++kt) {
      int k0 = kt * 4 + half * 2;
      v2f bf;
      bf.x = W1[(size_t)k0 * Hd + ht * 16 + lid];        // B[k][n] = W1[k][h]
      bf.y = W1[(size_t)(k0 + 1) * Hd + ht * 16 + lid];
      acc = __builtin_amdgcn_wmma_f32_16x16x4_f32(
          false, afrag[kt], false, bf, (short)0, acc, false, false);
    }
    #pragma unroll
    for (int v = 0; v < 8; ++v) {
      int mrow = v + half * 8;
      int hcol = ht * 16 + lid;
      float zz = acc[v] + b1[hcol];
      zz = fminf(fmaxf(zz, -40.f), 40.f);         // exp stays finite
      float e = __expf(2.f * zz);
      float r = __builtin_amdgcn_rcpf(e + 1.f);   // single v_rcp_f32
      s[wave][mrow][hcol] = (4.f * e) * r * r * w2[hcol];  // sech^2 * w2
    }
  }
  __syncthreads();

  // ---- GEMM2: g = s @ W1^T  (B[k][n] = W1[n][k], contiguous float2) ------
  for (int nt = 0; nt < Ncol / 16; ++nt) {
    v8f acc = {};
    for (int kt = 0; kt < Hd / 4; ++kt) {
      int k0 = kt * 4 + half * 2;
      v2f a, bf;
      a.x = s[wave][lid][k0];
      a.y = s[wave][lid][k0 + 1];
      int ng = nt * 16 + lid;
      bf.x = W1[(size_t)ng * Hd + k0];
      bf.y = W1[(size_t)ng * Hd + k0 + 1];
      acc = __builtin_amdgcn_wmma_f32_16x16x4_f32(
          false, a, false, bf, (short)0, acc, false, false);
    }
    #pragma unroll
    for (int v = 0; v < 8; ++v) {
      int mrow = v + half * 8;
      g[(size_t)(row0 + mrow) * Ncol + nt * 16 + lid] = acc[v];
    }
  }
}

// ---------------------------------------------------------------------------
extern "C" void kernel_launch(void* const* d_in, const int* in_sizes, int n_in,
                              void* d_out, int out_size, void* d_ws, size_t ws_size,
                              hipStream_t stream) {
  const float* x_raw = (const float*)d_in[0];  // [B,S,32]
  const float* A     = (const float*)d_in[1];  // [B,S,24,32]
  const float* b     = (const float*)d_in[2];  // [B,S,24]
  const float* W1    = (const float*)d_in[3];  // [32,256]
  const float* b1    = (const float*)d_in[4];  // [256]
  const float* w2    = (const float*)d_in[5];  // [256]
  float* out = (float*)d_out;

  const int P = in_sizes[0] / Ncol;            // flattened batch (16384)

  float* xfeas = (float*)d_ws;                 // [P,32]
  float* g     = xfeas + (size_t)P * Ncol;     // [P,32]

  // 1) anchor QP -> x_feas
  admm_kernel<Ncol + NSl, Mrow + NSl + Ncol, true>
      <<<P, 64, 0, stream>>>(x_raw, A, b, nullptr, xfeas);

  // 2) critic gradient via WMMA -> g   (P multiple of 32: 512 blocks)
  critic_grad_kernel<<<P / 32, 64, 0, stream>>>(xfeas, W1, b1, w2, g);

  // 3) LMO LP + Frank-Wolfe blend -> out
  admm_kernel<Ncol, Mrow + Ncol, false>
      <<<P, 64, 0, stream>>>(g, A, b, xfeas, out);
}